// Interaction_33457795236217
// MI455X (gfx1250) — compile-verified
//
#include <hip/hip_runtime.h>
#include <math.h>

// CDNA5 / gfx1250, wave32. fp32 WMMA: D(16x16) = A(16x4) * B(4x16) + C.
typedef float v2f __attribute__((ext_vector_type(2)));
typedef float v8f __attribute__((ext_vector_type(8)));

#define CUTOFF 4.5f

// ---------------- K1: normalize X, decompose into compact I(1)/A(3)/S(6) ----
__global__ void k1_decompose(const float* __restrict__ X, float* __restrict__ Xn,
                             float* __restrict__ cI, float* __restrict__ cA,
                             float* __restrict__ cS, int Np, int H) {
  int idx = blockIdx.x * blockDim.x + threadIdx.x;
  if (idx >= Np * H) return;
  int n = idx / H;
  int c = idx - n * H;
  const float* x = X + (size_t)idx * 9;
  float m[9];
  float ss = 0.f;
#pragma unroll
  for (int t = 0; t < 9; ++t) { m[t] = x[t]; ss += m[t] * m[t]; }
  float inv = 1.f / (ss + 1.f);
  float* xo = Xn + (size_t)idx * 9;
#pragma unroll
  for (int t = 0; t < 9; ++t) { m[t] *= inv; xo[t] = m[t]; }
  float tr = (m[0] + m[4] + m[8]) * (1.f / 3.f);
  cI[idx] = tr;
  float* ap = cA + ((size_t)n * 3) * H + c;
  ap[0]     = 0.5f * (m[1] - m[3]);
  ap[H]     = 0.5f * (m[2] - m[6]);
  ap[2 * H] = 0.5f * (m[5] - m[7]);
  float* sp = cS + ((size_t)n * 6) * H + c;
  sp[0]     = m[0] - tr;
  sp[H]     = 0.5f * (m[1] + m[3]);
  sp[2 * H] = 0.5f * (m[2] + m[6]);
  sp[3 * H] = m[4] - tr;
  sp[4 * H] = 0.5f * (m[5] + m[7]);
  sp[5 * H] = m[8] - tr;
}

// ---------------- weight pack: W[N,K] -> WMMA-B fragment layout --------------
// Wp[(k/4)*N*4 + n*4 + (k&3)] : each lane's (k, k+1) pair is one aligned b64,
// coalesced across lanes (lane = n mod 16).
__global__ void pack_w(const float* __restrict__ W, float* __restrict__ Wp,
                       int Nr, int Kc) {
  int idx = blockIdx.x * blockDim.x + threadIdx.x;
  if (idx >= Nr * Kc) return;
  int nrow = idx / Kc;
  int k = idx - nrow * Kc;
  Wp[((k >> 2) * Nr + nrow) * 4 + (k & 3)] = W[idx];
}

// ---------------- cosine cutoff factor per edge ------------------------------
__global__ void cutoff_kernel(const float* __restrict__ w, float* __restrict__ Cc,
                              int E) {
  int e = blockIdx.x * blockDim.x + threadIdx.x;
  if (e >= E) return;
  float x = w[e];
  float v = 0.5f * (__cosf(x * (3.14159265358979323846f / CUTOFF)) + 1.f);
  Cc[e] = (x < CUTOFF) ? v : 0.f;
}

// ---------------- WMMA fp32 GEMM: C[M,N] = A[M,K] @ W^T (+bias,silu,rowscale)
// block: 128 thr = 4 waves; block tile 32 rows x 128 cols; each wave owns a
// 32x32 register tile = 2x2 WMMA subtiles (2 A-frags x 2 B-frags -> 4 WMMA).
// Full-K A panel staged in LDS (K in {64,128,256}, power of two).
__global__ __launch_bounds__(128) void wmma_gemm_f32(
    const float* __restrict__ A, const float* __restrict__ Wp,
    const float* __restrict__ bias, const float* __restrict__ rowscale,
    float* __restrict__ C, int M, int K, int N, int act) {
  __shared__ float As[32 * 260];
  const int Kp = K + 4;                 // 16B-aligned rows, bank-spread stride
  const int row0 = blockIdx.x * 32;
  const int kshift = 31 - __clz(K);
  const int kmask = K - 1;
  // stage 32 x K panel with coalesced float4 loads
  const int quads = (K << 5) >> 2;
  for (int i = threadIdx.x; i < quads; i += 128) {
    int e4 = i << 2;
    int r = e4 >> kshift;
    int k = e4 & kmask;
    int row = row0 + r;
    float4 v = make_float4(0.f, 0.f, 0.f, 0.f);
    if (row < M) v = *(const float4*)(A + row * K + k);
    *(float4*)(&As[r * Kp + k]) = v;
  }
  __syncthreads();

  const int lane = threadIdx.x & 31;
  const int w = threadIdx.x >> 5;       // 0..3 -> 32-col slice
  const int laneM = lane & 15;          // A: M / B: N / D: N index
  const int kk = lane >> 4;             // half-wave selects K pair, D row +8
  const int nBase = blockIdx.y * 128 + w * 32;
  int nc0 = nBase + laneM;
  int nc1 = nc0 + 16;
  int ncl0 = (nc0 < N) ? nc0 : (N - 1);
  int ncl1 = (nc1 < N) ? nc1 : (N - 1);

  v8f acc00 = {}, acc01 = {}, acc10 = {}, acc11 = {};
  const float* Ap0 = As + laneM * Kp + 2 * kk;
  const float* Ap1 = As + (16 + laneM) * Kp + 2 * kk;
#pragma unroll 4
  for (int k0 = 0; k0 < K; k0 += 4) {
    v2f a0 = *(const v2f*)(Ap0 + k0);
    v2f a1 = *(const v2f*)(Ap1 + k0);
    const float* wq = Wp + ((k0 >> 2) * N) * 4 + 2 * kk;
    v2f b0 = *(const v2f*)(wq + ncl0 * 4);
    v2f b1 = *(const v2f*)(wq + ncl1 * 4);
    // (neg_a, A, neg_b, B, c_mod, C, reuse_a, reuse_b)
    acc00 = __builtin_amdgcn_wmma_f32_16x16x4_f32(false, a0, false, b0, (short)0,
                                                  acc00, false, false);
    acc01 = __builtin_amdgcn_wmma_f32_16x16x4_f32(false, a0, false, b1, (short)0,
                                                  acc01, false, false);
    acc10 = __builtin_amdgcn_wmma_f32_16x16x4_f32(false, a1, false, b0, (short)0,
                                                  acc10, false, false);
    acc11 = __builtin_amdgcn_wmma_f32_16x16x4_f32(false, a1, false, b1, (short)0,
                                                  acc11, false, false);
  }

  float bs0 = (bias && nc0 < N) ? bias[nc0] : 0.f;
  float bs1 = (bias && nc1 < N) ? bias[nc1] : 0.f;
#pragma unroll
  for (int r = 0; r < 8; ++r) {
    int rowA = row0 + r + 8 * kk;        // D layout: VGPR r -> M = r + 8*kk
    int rowB = rowA + 16;
    float rsA = 1.f, rsB = 1.f;
    if (rowscale) {
      if (rowA < M) rsA = rowscale[rowA];
      if (rowB < M) rsB = rowscale[rowB];
    }
    if (rowA < M) {
      if (nc0 < N) {
        float v = acc00[r] + bs0;
        if (act) v = v / (1.f + __expf(-v));
        C[rowA * N + nc0] = v * rsA;
      }
      if (nc1 < N) {
        float v = acc01[r] + bs1;
        if (act) v = v / (1.f + __expf(-v));
        C[rowA * N + nc1] = v * rsA;
      }
    }
    if (rowB < M) {
      if (nc0 < N) {
        float v = acc10[r] + bs0;
        if (act) v = v / (1.f + __expf(-v));
        C[rowB * N + nc0] = v * rsB;
      }
      if (nc1 < N) {
        float v = acc11[r] + bs1;
        if (act) v = v / (1.f + __expf(-v));
        C[rowB * N + nc1] = v * rsB;
      }
    }
  }
}

// ---------------- message passing: compact scatter with f32 atomics ----------
// msg layout [n, 13, H]: slot0 = I, 1..3 = A, 4..9 = S, 10..12 = sum of edge
// factors (needed because chan_linear bias lands on all 9 matrix entries).
__global__ void mp_kernel(const int* __restrict__ EI, const float* __restrict__ ea,
                          const float* __restrict__ i1, const float* __restrict__ a1,
                          const float* __restrict__ s1, float* __restrict__ msg,
                          int E, int H) {
  int g = blockIdx.x * blockDim.x + threadIdx.x;
  if (g >= E * H) return;
  int e = g / H;
  int c = g - e * H;
  int dst = EI[e];
  int src = EI[E + e];
  const float* ep = ea + (size_t)e * 3 * H + c * 3;
  float f0 = ep[0], f1 = ep[1], f2 = ep[2];
  float* mb = msg + (size_t)dst * 13 * H + c;
  atomicAdd(mb, f0 * i1[(size_t)src * H + c]);
  const float* ap = a1 + (size_t)src * 3 * H + c;
  atomicAdd(mb + 1 * H, f1 * ap[0]);
  atomicAdd(mb + 2 * H, f1 * ap[H]);
  atomicAdd(mb + 3 * H, f1 * ap[2 * H]);
  const float* sp = s1 + (size_t)src * 6 * H + c;
#pragma unroll
  for (int t = 0; t < 6; ++t) atomicAdd(mb + (4 + t) * H, f2 * sp[t * H]);
  atomicAdd(mb + 10 * H, f0);
  atomicAdd(mb + 11 * H, f1);
  atomicAdd(mb + 12 * H, f2);
}

// ---------------- K2: NF = M@Y + Y@M, renormalize, re-decompose --------------
__global__ void k2_second(const float* __restrict__ msg, const float* __restrict__ i1,
                          const float* __restrict__ a1, const float* __restrict__ s1,
                          const float* __restrict__ b1, float* __restrict__ cI2,
                          float* __restrict__ cA2, float* __restrict__ cS2,
                          int Np, int H) {
  int idx = blockIdx.x * blockDim.x + threadIdx.x;
  if (idx >= Np * H) return;
  int n = idx / H;
  int c = idx - n * H;
  float b0 = b1[c], bA = b1[H + c], bS = b1[2 * H + c];
  float B1 = b0 + bA + bS;
  float yi = i1[idx];
  const float* ap = a1 + (size_t)n * 3 * H + c;
  float ya0 = ap[0], ya1 = ap[H], ya2 = ap[2 * H];
  const float* sp = s1 + (size_t)n * 6 * H + c;
  float ys0 = sp[0], ys1 = sp[H], ys2 = sp[2 * H];
  float ys3 = sp[3 * H], ys4 = sp[4 * H], ys5 = sp[5 * H];
  float Y[3][3] = {
      { yi + ys0 + B1,  ya0 + ys1 + B1,  ya1 + ys2 + B1},
      {-ya0 + ys1 + B1,  yi + ys3 + B1,  ya2 + ys4 + B1},
      {-ya1 + ys2 + B1, -ya2 + ys4 + B1,  yi + ys5 + B1}};
  const float* q = msg + (size_t)n * 13 * H + c;
  float mi = q[0];
  float ma0 = q[H], ma1 = q[2 * H], ma2 = q[3 * H];
  float ms0 = q[4 * H], ms1 = q[5 * H], ms2 = q[6 * H];
  float ms3 = q[7 * H], ms4 = q[8 * H], ms5 = q[9 * H];
  float F0 = q[10 * H], F1 = q[11 * H], F2 = q[12 * H];
  float Mb = F0 * b0 + F1 * bA + F2 * bS;
  float Mm[3][3] = {
      { mi + ms0 + Mb,  ma0 + ms1 + Mb,  ma1 + ms2 + Mb},
      {-ma0 + ms1 + Mb,  mi + ms3 + Mb,  ma2 + ms4 + Mb},
      {-ma1 + ms2 + Mb, -ma2 + ms4 + Mb,  mi + ms5 + Mb}};
  float NF[3][3];
  float ss = 0.f;
#pragma unroll
  for (int i = 0; i < 3; ++i)
#pragma unroll
    for (int j = 0; j < 3; ++j) {
      float v = 0.f;
#pragma unroll
      for (int k = 0; k < 3; ++k) v += Mm[i][k] * Y[k][j] + Y[i][k] * Mm[k][j];
      NF[i][j] = v;
      ss += v * v;
    }
  float inv = 1.f / (ss + 1.f);   // tensor_norm(I+A+S) == ||NF||^2
  float tr = (NF[0][0] + NF[1][1] + NF[2][2]) * (1.f / 3.f);
  cI2[idx] = tr * inv;
  float* oa = cA2 + (size_t)n * 3 * H + c;
  oa[0]     = 0.5f * (NF[0][1] - NF[1][0]) * inv;
  oa[H]     = 0.5f * (NF[0][2] - NF[2][0]) * inv;
  oa[2 * H] = 0.5f * (NF[1][2] - NF[2][1]) * inv;
  float* os = cS2 + (size_t)n * 6 * H + c;
  os[0]     = (NF[0][0] - tr) * inv;
  os[H]     = 0.5f * (NF[0][1] + NF[1][0]) * inv;
  os[2 * H] = 0.5f * (NF[0][2] + NF[2][0]) * inv;
  os[3 * H] = (NF[1][1] - tr) * inv;
  os[4 * H] = 0.5f * (NF[1][2] + NF[2][1]) * inv;
  os[5 * H] = (NF[2][2] - tr) * inv;
}

// ---------------- K3: out = Xn + dX + dX@dX ---------------------------------
__global__ void k3_final(const float* __restrict__ Xn, const float* __restrict__ i3,
                         const float* __restrict__ a3, const float* __restrict__ s3,
                         const float* __restrict__ b3, float* __restrict__ out,
                         int Np, int H) {
  int idx = blockIdx.x * blockDim.x + threadIdx.x;
  if (idx >= Np * H) return;
  int n = idx / H;
  int c = idx - n * H;
  float B3 = b3[c] + b3[H + c] + b3[2 * H + c];
  float di = i3[idx];
  const float* ap = a3 + (size_t)n * 3 * H + c;
  float a0 = ap[0], a1v = ap[H], a2 = ap[2 * H];
  const float* sp = s3 + (size_t)n * 6 * H + c;
  float s0 = sp[0], s1v = sp[H], s2 = sp[2 * H];
  float s3v = sp[3 * H], s4 = sp[4 * H], s5 = sp[5 * H];
  float D[3][3] = {
      { di + s0 + B3,   a0 + s1v + B3,  a1v + s2 + B3},
      {-a0 + s1v + B3,  di + s3v + B3,  a2 + s4 + B3},
      {-a1v + s2 + B3, -a2 + s4 + B3,   di + s5 + B3}};
  const float* xp = Xn + (size_t)idx * 9;
  float* op = out + (size_t)idx * 9;
#pragma unroll
  for (int i = 0; i < 3; ++i)
#pragma unroll
    for (int j = 0; j < 3; ++j) {
      float dd = 0.f;
#pragma unroll
      for (int k = 0; k < 3; ++k) dd += D[i][k] * D[k][j];
      op[i * 3 + j] = xp[i * 3 + j] + D[i][j] + dd;
    }
}

extern "C" void kernel_launch(void* const* d_in, const int* in_sizes, int n_in,
                              void* d_out, int out_size, void* d_ws, size_t ws_size,
                              hipStream_t stream) {
  const float* X   = (const float*)d_in[0];
  const int*   EI  = (const int*)d_in[1];
  const float* EW  = (const float*)d_in[2];
  const float* EA  = (const float*)d_in[3];
  const float* W1  = (const float*)d_in[4];
  const float* b1  = (const float*)d_in[5];
  const float* W2a = (const float*)d_in[6];
  const float* b2a = (const float*)d_in[7];
  const float* W2b = (const float*)d_in[8];
  const float* b2b = (const float*)d_in[9];
  const float* W2c = (const float*)d_in[10];
  const float* b2c = (const float*)d_in[11];
  const float* W3  = (const float*)d_in[12];
  const float* b3  = (const float*)d_in[13];
  float* out = (float*)d_out;

  const int H  = in_sizes[5] / 3;          // 128
  const int Np = in_sizes[0] / (9 * H);    // 20001
  const int E  = in_sizes[2];              // 200000
  const int R  = in_sizes[3] / E;          // 64

  // workspace layout (floats), with lifetime-based aliasing (~945 MB total)
  float* ws = (float*)d_ws;
  size_t NH = (size_t)Np * H;
  size_t cset = 10 * NH;
  size_t c1sz = cset > (size_t)E * H ? cset : (size_t)E * H;
  size_t h2sz = (size_t)E * 2 * H > cset ? (size_t)E * 2 * H : cset;
  auto al = [](size_t v) { return (v + 15) & ~(size_t)15; };
  size_t o_Xn = 0;
  size_t o_c1 = al(o_Xn + NH * 9);     // c1 compacts -> h1 -> c2 compacts
  size_t o_s1 = al(o_c1 + c1sz);       // stage-1 chan_linear outputs
  size_t o_h2 = al(o_s1 + cset);       // h2 -> stage-2 chan_linear outputs
  size_t o_ea = al(o_h2 + h2sz);       // edge factors [E,3H]
  size_t o_ms = al(o_ea + (size_t)E * 3 * H);   // msg [Np,13,H]
  size_t o_wt = al(o_ms + 13 * NH);    // packed-weight scratch
  size_t o_cc = al(o_wt + (size_t)3 * H * 2 * H);  // cutoff factors [E]
  float* Xn  = ws + o_Xn;
  float* cI  = ws + o_c1;  float* cA = cI + NH;  float* cS = cI + 4 * NH;
  float* h1  = ws + o_c1;
  float* i1  = ws + o_s1;  float* a1 = i1 + NH;  float* s1 = i1 + 4 * NH;
  float* h2  = ws + o_h2;
  float* i3  = ws + o_h2;  float* a3 = i3 + NH;  float* s3p = i3 + 4 * NH;
  float* ea  = ws + o_ea;
  float* msg = ws + o_ms;
  float* Wp  = ws + o_wt;
  float* Cc  = ws + o_cc;

  int NHi = Np * H;

  auto launch_gemm = [&](const float* Amat, const float* Wmat, const float* bias,
                         const float* rowscale, float* Co, int M, int K, int Nn,
                         int act) {
    int tN = Nn * K;
    pack_w<<<(tN + 255) / 256, 256, 0, stream>>>(Wmat, Wp, Nn, K);
    dim3 g((M + 31) / 32, (Nn + 127) / 128);
    wmma_gemm_f32<<<g, 128, 0, stream>>>(Amat, Wp, bias, rowscale, Co, M, K, Nn,
                                         act);
  };

  // 1) normalize + decompose
  k1_decompose<<<(NHi + 255) / 256, 256, 0, stream>>>(X, Xn, cI, cA, cS, Np, H);

  // 2) chan_linear stage 1 (biases deferred to reconstruction)
  launch_gemm(cI, W1 + 0 * H * H, nullptr, nullptr, i1, Np,     H, H, 0);
  launch_gemm(cA, W1 + 1 * H * H, nullptr, nullptr, a1, 3 * Np, H, H, 0);
  launch_gemm(cS, W1 + 2 * H * H, nullptr, nullptr, s1, 6 * Np, H, H, 0);

  // 3) edge MLP (silu), cutoff folded into last epilogue
  cutoff_kernel<<<(E + 255) / 256, 256, 0, stream>>>(EW, Cc, E);
  launch_gemm(EA, W2a, b2a, nullptr, h1, E, R,     H,     1);
  launch_gemm(h1, W2b, b2b, nullptr, h2, E, H,     2 * H, 1);
  launch_gemm(h2, W2c, b2c, Cc,      ea, E, 2 * H, 3 * H, 1);

  // 4) message passing (compact, atomic scatter)
  hipMemsetAsync(msg, 0, 13 * NH * sizeof(float), stream);
  mp_kernel<<<(int)(((size_t)E * H + 255) / 256), 256, 0, stream>>>(EI, ea, i1, a1,
                                                                    s1, msg, E, H);

  // 5) NF = M@Y + Y@M, renormalize, re-decompose (writes c2 into c1 region)
  k2_second<<<(NHi + 255) / 256, 256, 0, stream>>>(msg, i1, a1, s1, b1, cI, cA, cS,
                                                   Np, H);

  // 6) chan_linear stage 2
  launch_gemm(cI, W3 + 0 * H * H, nullptr, nullptr, i3,  Np,     H, H, 0);
  launch_gemm(cA, W3 + 1 * H * H, nullptr, nullptr, a3,  3 * Np, H, H, 0);
  launch_gemm(cS, W3 + 2 * H * H, nullptr, nullptr, s3p, 6 * Np, H, H, 0);

  // 7) out = Xn + dX + dX@dX
  k3_final<<<(NHi + 255) / 256, 256, 0, stream>>>(Xn, i3, a3, s3p, b3, out, Np, H);
}